// SingleHeadAttention_66984309948497
// MI455X (gfx1250) — compile-verified
//
#include <hip/hip_runtime.h>
#include <hip/hip_bf16.h>
#include <math.h>

#define N_NODES 10000
#define N_EDGES 320000
#define D 128
#define HASH_CAP (1u << 20)
#define HASH_MASK (HASH_CAP - 1u)
#define EMPTY_KEY 0xFFFFFFFFu

typedef __attribute__((ext_vector_type(2))) float v2f;
typedef __attribute__((ext_vector_type(8))) float v8f;

__device__ __forceinline__ unsigned hash_u32(unsigned k) {
    k *= 0x9E3779B1u;
    k ^= k >> 16;
    k *= 0x85EBCA77u;
    k ^= k >> 13;
    return k;
}

// ---------------------------------------------------------------------------
// K0: zero the output accumulator, denom, and reset the hash table.
// ---------------------------------------------------------------------------
__global__ void k_init(float* __restrict__ out, float* __restrict__ denom,
                       unsigned* __restrict__ hkeys, float* __restrict__ hvals) {
    const int tid = blockIdx.x * blockDim.x + threadIdx.x;
    const int stride = gridDim.x * blockDim.x;
    for (int i = tid; i < N_NODES * D; i += stride) out[i] = 0.0f;
    for (int i = tid; i < N_NODES; i += stride) denom[i] = 0.0f;
    for (unsigned i = tid; i < HASH_CAP; i += stride) {
        hkeys[i] = EMPTY_KEY;
        hvals[i] = 0.0f;
    }
}

// ---------------------------------------------------------------------------
// K1: ak = X @ W via V_WMMA_F32_16X16X4_F32, one wave32 per 16x16 tile.
// M = 10000 (625 tiles), N = 128 (8 tiles), K = 128 (32 wmma steps of 4).
// A frag layout (16x4 f32): lanes 0-15 -> K=0,1 ; lanes 16-31 -> K=2,3 (row = lane&15)
// B frag layout (4x16 f32): mirrors A with col = lane&15.
// D layout: VGPR v -> row v + 8*(lane>>4), col lane&15.
// ---------------------------------------------------------------------------
__global__ __launch_bounds__(32) void k_gemm(const float* __restrict__ X,
                                             const float* __restrict__ W,
                                             float* __restrict__ ak) {
    const int mt   = blockIdx.x >> 3;   // 0..624
    const int nt   = blockIdx.x & 7;    // 0..7
    const int lane = threadIdx.x;       // 0..31, full EXEC
    const int half = lane >> 4;         // 0|1
    const int l15  = lane & 15;
    const int row  = mt * 16 + l15;
    const int col  = nt * 16 + l15;
    const int koff = half * 2;

    const float* __restrict__ xrow = X + (size_t)row * D;
    v8f acc = {};
#pragma unroll 4
    for (int k = 0; k < D; k += 4) {
        v2f a, b;
        a.x = xrow[k + koff];
        a.y = xrow[k + koff + 1];
        b.x = W[(size_t)(k + koff) * D + col];
        b.y = W[(size_t)(k + koff + 1) * D + col];
        acc = __builtin_amdgcn_wmma_f32_16x16x4_f32(
            /*neg_a=*/false, a, /*neg_b=*/false, b,
            /*c_mod=*/(short)0, acc, /*reuse_a=*/false, /*reuse_b=*/false);
    }
#pragma unroll
    for (int v = 0; v < 8; ++v) {
        const int m = mt * 16 + v + half * 8;
        ak[(size_t)m * D + nt * 16 + l15] = acc[v];
    }
}

// ---------------------------------------------------------------------------
// K2: per-node left/right attention scalars. One wave per node, float4 loads,
// wave32 shuffle reduction.
// ---------------------------------------------------------------------------
__global__ __launch_bounds__(256) void k_lr(const float* __restrict__ ak,
                                            const float* __restrict__ attn,
                                            float* __restrict__ left,
                                            float* __restrict__ right) {
    const int wave = threadIdx.x >> 5;
    const int lane = threadIdx.x & 31;
    const int node = blockIdx.x * 8 + wave;
    if (node >= N_NODES) return;

    const float4 v = ((const float4*)(ak + (size_t)node * D))[lane];
    const float4 l = ((const float4*)attn)[lane];        // attn[0:128]
    const float4 r = ((const float4*)(attn + D))[lane];  // attn[128:256]
    float dl = v.x * l.x + v.y * l.y + v.z * l.z + v.w * l.w;
    float dr = v.x * r.x + v.y * r.y + v.z * r.z + v.w * r.w;
#pragma unroll
    for (int off = 16; off > 0; off >>= 1) {
        dl += __shfl_xor(dl, off, 32);
        dr += __shfl_xor(dr, off, 32);
    }
    if (lane == 0) {
        left[node]  = dl;
        right[node] = dr;
    }
}

// ---------------------------------------------------------------------------
// K3: score_e = exp(left[src] + right[dst]); denom[src] += score.
// ---------------------------------------------------------------------------
__global__ void k_scores(const int* __restrict__ el,
                         const float* __restrict__ left,
                         const float* __restrict__ right,
                         float* __restrict__ scores,
                         float* __restrict__ denom) {
    const int e = blockIdx.x * blockDim.x + threadIdx.x;
    if (e >= N_EDGES) return;
    const int s = el[2 * e + 0];
    const int d = el[2 * e + 1];
    const float sc = expf(left[s] + right[d]);
    scores[e] = sc;
    atomicAdd(&denom[s], sc);
}

// ---------------------------------------------------------------------------
// K4: coeff = score/denom[src]; hash-accumulate at key (src,dst)
// (equivalent to dense.at[src,dst].add(coeff), duplicates included).
// ---------------------------------------------------------------------------
__global__ void k_insert(const int* __restrict__ el,
                         const float* __restrict__ scores,
                         const float* __restrict__ denom,
                         unsigned* __restrict__ hkeys,
                         float* __restrict__ hvals) {
    const int e = blockIdx.x * blockDim.x + threadIdx.x;
    if (e >= N_EDGES) return;
    const int s = el[2 * e + 0];
    const int d = el[2 * e + 1];
    const float dn = denom[s];
    const float c  = (dn == 0.0f) ? 0.0f : scores[e] / dn;
    const unsigned key = (unsigned)s * (unsigned)N_NODES + (unsigned)d;  // < 1e8
    unsigned h = hash_u32(key) & HASH_MASK;
    for (;;) {
        const unsigned old = atomicCAS(&hkeys[h], EMPTY_KEY, key);
        if (old == EMPTY_KEY || old == key) {
            atomicAdd(&hvals[h], c);
            break;
        }
        h = (h + 1u) & HASH_MASK;
    }
}

// ---------------------------------------------------------------------------
// K5: coeff_e = dense[dst_e, src_e] (reverse-edge lookup);
//     out[src_e] += ak[dst_e] * coeff_e.  One wave per edge, float4 per lane.
// ---------------------------------------------------------------------------
__global__ __launch_bounds__(256) void k_scatter(const int* __restrict__ el,
                                                 const unsigned* __restrict__ hkeys,
                                                 const float* __restrict__ hvals,
                                                 const float* __restrict__ ak,
                                                 float* __restrict__ out) {
    const int wave = threadIdx.x >> 5;
    const int lane = threadIdx.x & 31;
    const int e = blockIdx.x * 8 + wave;
    if (e >= N_EDGES) return;
    const int s = el[2 * e + 0];
    const int d = el[2 * e + 1];
    const unsigned key = (unsigned)d * (unsigned)N_NODES + (unsigned)s;  // dense[dst, src]
    float coeff = 0.0f;
    unsigned h = hash_u32(key) & HASH_MASK;
    for (;;) {
        const unsigned kk = hkeys[h];
        if (kk == key) { coeff = hvals[h]; break; }
        if (kk == EMPTY_KEY) break;  // reverse edge absent -> 0
        h = (h + 1u) & HASH_MASK;
    }
    if (coeff != 0.0f) {
        const float4 v = ((const float4*)(ak + (size_t)d * D))[lane];
        float* o = out + (size_t)s * D + lane * 4;
        atomicAdd(o + 0, v.x * coeff);
        atomicAdd(o + 1, v.y * coeff);
        atomicAdd(o + 2, v.z * coeff);
        atomicAdd(o + 3, v.w * coeff);
    }
}

// ---------------------------------------------------------------------------
// K6: out = relu(out + bias)
// ---------------------------------------------------------------------------
__global__ void k_final(float* __restrict__ out, const float* __restrict__ bias) {
    const int i = blockIdx.x * blockDim.x + threadIdx.x;
    if (i >= N_NODES * D) return;
    const float x = out[i] + bias[i & (D - 1)];
    out[i] = x > 0.0f ? x : 0.0f;
}

extern "C" void kernel_launch(void* const* d_in, const int* in_sizes, int n_in,
                              void* d_out, int out_size, void* d_ws, size_t ws_size,
                              hipStream_t stream) {
    const float* X    = (const float*)d_in[0];  // node_features (10000,128)
    const int*   el   = (const int*)d_in[1];    // edge_list (320000,2)
    const float* W    = (const float*)d_in[2];  // kernel (128,128)
    const float* attn = (const float*)d_in[3];  // attention_kernel (256,1)
    const float* bias = (const float*)d_in[4];  // bias (128,)
    float* out = (float*)d_out;                 // (10000,128)

    // Workspace carve-out (256B aligned): ~16 MB total.
    char* ws = (char*)d_ws;
    size_t off = 0;
    auto carve = [&](size_t bytes) -> char* {
        char* p = ws + off;
        off = (off + bytes + 255) & ~(size_t)255;
        return p;
    };
    float*    ak     = (float*)carve((size_t)N_NODES * D * sizeof(float));
    float*    left   = (float*)carve((size_t)N_NODES * sizeof(float));
    float*    right  = (float*)carve((size_t)N_NODES * sizeof(float));
    float*    denom  = (float*)carve((size_t)N_NODES * sizeof(float));
    float*    scores = (float*)carve((size_t)N_EDGES * sizeof(float));
    unsigned* hkeys  = (unsigned*)carve((size_t)HASH_CAP * sizeof(unsigned));
    float*    hvals  = (float*)carve((size_t)HASH_CAP * sizeof(float));

    k_init<<<2048, 256, 0, stream>>>(out, denom, hkeys, hvals);
    k_gemm<<<(N_NODES / 16) * (D / 16), 32, 0, stream>>>(X, W, ak);
    k_lr<<<(N_NODES + 7) / 8, 256, 0, stream>>>(ak, attn, left, right);
    k_scores<<<(N_EDGES + 255) / 256, 256, 0, stream>>>(el, left, right, scores, denom);
    k_insert<<<(N_EDGES + 255) / 256, 256, 0, stream>>>(el, scores, denom, hkeys, hvals);
    k_scatter<<<(N_EDGES + 7) / 8, 256, 0, stream>>>(el, hkeys, hvals, ak, out);
    k_final<<<(N_NODES * D + 255) / 256, 256, 0, stream>>>(out, bias);
}